// RBFLoss_1958505087543
// MI455X (gfx1250) — compile-verified
//
#include <hip/hip_runtime.h>

// ---------------------------------------------------------------------------
// RBF loss, single-pass streaming reduction. Memory floor: 4M rows x 44 B
// ~= 184 MB @ 23.3 TB/s ~= 8 us. Designed to be issue-rate lean (~2.3
// instructions/row) so the wave front actually sustains the HBM roofline.
//
// Per wave iteration: 32 rows = two 16x12 A-tiles (K=10 zero-padded to 12)
// multiplied against the constant TABLE^T via 6x V_WMMA_F32_16X16X4_F32
// (two interleaved accumulator chains). B-fragments are compile-time
// constants -> the table costs zero bandwidth. Pad K-columns need no A
// zeroing: their B-fragment entries are 0, so garbage*0 == 0.
//   norm_m = 1 + |x2[t]|^2 - 2*D[m][t] * rsqrt(|x1_m|^2)
//   loss  += 1 - exp(-norm_m)
// Tail runs on all 32 lanes (lanes 0-15 -> tile A rows, 16-31 -> tile B).
// Reduction: shfl_xor tree -> one global_atomic_add_f32 per wave.
// ---------------------------------------------------------------------------

typedef float v2f __attribute__((ext_vector_type(2)));
typedef float v8f __attribute__((ext_vector_type(8)));

__constant__ float TBL[10][10] = {
  {0.f, 0.333f, 0.333f, 0.333f, 0.333f, 0.333f, 0.333f, 0.333f, 0.333f, 0.333f},
  {1.f, 0.f, 0.f, 0.f, 0.f, 0.f, 0.f, 0.f, 0.f, 0.f},
  {0.f, -0.118f, 0.943f, -0.118f, -0.118f, -0.118f, -0.118f, -0.118f, -0.118f, -0.118f},
  {0.f, -0.134f, 1.78e-16f, 0.935f, -0.134f, -0.134f, -0.134f, -0.134f, -0.134f, -0.134f},
  {0.f, -0.154f, 1.999e-16f, 3.997e-16f, 0.926f, -0.154f, -0.154f, -0.154f, -0.154f, -0.154f},
  {0.f, -0.183f, 1.824e-16f, 3.953e-16f, 6.081e-17f, 0.913f, -0.183f, -0.183f, -0.183f, -0.183f},
  {0.f, -0.224f, 1.738e-16f, 4.469e-16f, 7.448e-17f, -9.93e-17f, 0.894f, -0.224f, -0.224f, -0.224f},
  {0.f, -0.289f, 2.134e-16f, 4.914e-16f, 6.41e-17f, -1.282e-16f, 1.282e-16f, 0.866f, -0.289f, -0.289f},
  {0.f, -0.408f, 3.108e-16f, 7.576e-16f, 9.712e-17f, -1.165e-16f, 2.331e-16f, -3.885e-17f, 0.816f, -0.408f},
  {0.f, -0.707f, 5.103e-16f, 1.178e-15f, 1.963e-16f, -1.57e-16f, 3.14e-16f, -3.925e-17f, -2.355e-16f, 0.707f},
};

__global__ void rbf_zero_out(float* out) {
  if (blockIdx.x == 0 && threadIdx.x == 0) out[0] = 0.0f;
}

__device__ __forceinline__ v2f nt_load2(const float* p) {
  return __builtin_nontemporal_load(reinterpret_cast<const v2f*>(p));
}

__global__ __launch_bounds__(256) void rbf_loss_wmma(
    const float* __restrict__ x1,
    const int*   __restrict__ tgt,
    float*       __restrict__ out,
    int nRows, int nPairs, float invB)
{
  // Per-wave pair of 16x16 D tiles in LDS (stride 17: bank-conflict free).
  __shared__ float sD[8][2 * 16 * 17];
  __shared__ float s2[16];

  const int tid  = threadIdx.x;
  const int lane = tid & 31;
  const int wid  = tid >> 5;          // 8 wave32 per 256-thread block
  const int n    = lane & 15;
  const int half = lane >> 4;

  // One-time: |TABLE[t]|^2 into LDS.
  if (tid < 16) {
    float s = 0.0f;
    if (tid < 10) {
      #pragma unroll
      for (int k = 0; k < 10; ++k) s = fmaf(TBL[tid][k], TBL[tid][k], s);
    }
    s2[tid] = s;
  }
  __syncthreads();

  // Constant B-fragments: B[k][n] = TABLE[n][k] (TABLE^T), zero-padded.
  // f32 4x16 layout: VGPR0 holds K = k0+2*half, VGPR1 holds K = k0+2*half+1.
  v2f bf[3];
  #pragma unroll
  for (int s = 0; s < 3; ++s) {
    const int k = 4 * s + 2 * half;
    bf[s].x = (n < 10 && k     < 10) ? TBL[n][k]     : 0.0f;
    bf[s].y = (n < 10 && k + 1 < 10) ? TBL[n][k + 1] : 0.0f;
  }

  const int wavesPerBlock = blockDim.x >> 5;
  const int gWave   = blockIdx.x * wavesPerBlock + wid;
  const int wStride = gridDim.x * wavesPerBlock;

  const unsigned kBase   = 2u * (unsigned)half;   // slice-0 column for this lane
  const unsigned k2Delta = half ? 6u : 8u;        // slice-2: half=1 dups k=8,9 (B entries are 0)
  const int      rowMax  = nRows - 1;

  float accLoss = 0.0f;

  for (int p = gWave; p < nPairs; p += wStride) {
    const int rowA = p * 32 + n;
    const int rowB = rowA + 16;

    // Prefetch next grid-stride pair (streaming, speculative).
    {
      const long long nr = (long long)(p + wStride) * 32 + n;
      if (nr < nRows) __builtin_prefetch(x1 + nr * 10, 0, 1);
    }

    // Clamped rows (only differs from rowA/rowB on the final partial pair).
    const unsigned rA = (unsigned)((rowA < nRows) ? rowA : rowMax);
    const unsigned rB = (unsigned)((rowB < nRows) ? rowB : rowMax);
    const unsigned iA = rA * 10u + kBase;
    const unsigned iB = rB * 10u + kBase;

    // 6x NT global_load_b64: all x1 traffic for 32 rows (pad dups coalesce).
    const v2f a0 = nt_load2(x1 + iA);
    const v2f a1 = nt_load2(x1 + iA + 4u);
    const v2f a2 = nt_load2(x1 + iA + k2Delta);
    const v2f g0 = nt_load2(x1 + iB);
    const v2f g1 = nt_load2(x1 + iB + 4u);
    const v2f g2 = nt_load2(x1 + iB + k2Delta);

    // Each lane finishes one row: lanes 0-15 -> tile A, 16-31 -> tile B.
    const unsigned rSel   = half ? rB : rA;
    const int      rowSel = half ? rowB : rowA;
    const int      t      = tgt[rSel];

    // |x1_row|^2 partials; mask the duplicated slice on upper-half lanes.
    float pa = fmaf(a0.x, a0.x, a0.y * a0.y);
    pa = fmaf(a1.x, a1.x, fmaf(a1.y, a1.y, pa));
    pa += half ? 0.0f : fmaf(a2.x, a2.x, a2.y * a2.y);
    float pb = fmaf(g0.x, g0.x, g0.y * g0.y);
    pb = fmaf(g1.x, g1.x, fmaf(g1.y, g1.y, pb));
    pb += half ? 0.0f : fmaf(g2.x, g2.x, g2.y * g2.y);
    const float ssA = pa + __shfl_xor(pa, 16, 32);
    const float ssB = pb + __shfl_xor(pb, 16, 32);
    const float ss  = half ? ssB : ssA;

    // Two interleaved WMMA accumulator chains (co-executable).
    v8f dA = {0.f, 0.f, 0.f, 0.f, 0.f, 0.f, 0.f, 0.f};
    v8f dB = {0.f, 0.f, 0.f, 0.f, 0.f, 0.f, 0.f, 0.f};
    dA = __builtin_amdgcn_wmma_f32_16x16x4_f32(false, a0, false, bf[0], (short)0, dA, false, false);
    dB = __builtin_amdgcn_wmma_f32_16x16x4_f32(false, g0, false, bf[0], (short)0, dB, false, false);
    dA = __builtin_amdgcn_wmma_f32_16x16x4_f32(false, a1, false, bf[1], (short)0, dA, false, false);
    dB = __builtin_amdgcn_wmma_f32_16x16x4_f32(false, g1, false, bf[1], (short)0, dB, false, false);
    dA = __builtin_amdgcn_wmma_f32_16x16x4_f32(false, a2, false, bf[2], (short)0, dA, false, false);
    dB = __builtin_amdgcn_wmma_f32_16x16x4_f32(false, g2, false, bf[2], (short)0, dB, false, false);

    // Spill both D tiles: D[r + 8*half][n] (pairs fuse to ds_store_2addr).
    #pragma unroll
    for (int r = 0; r < 8; ++r) {
      sD[wid][(r + 8 * half) * 17 + n]       = dA[r];
      sD[wid][272 + (r + 8 * half) * 17 + n] = dB[r];
    }
    __builtin_amdgcn_wave_barrier();   // LDS is in-order per wave; keep program order

    // Full-width tail: each lane gathers its row's target column.
    const float dt   = sD[wid][half * 272 + n * 17 + t];
    const float norm = 1.0f + s2[t] - (dt + dt) * rsqrtf(ss);
    const float li   = 1.0f - __expf(-norm);       // GAMMA = 1
    accLoss += (rowSel < nRows) ? li : 0.0f;
    __builtin_amdgcn_wave_barrier();
  }

  // Wave reduction, then one f32 atomic per wave.
  #pragma unroll
  for (int off = 16; off > 0; off >>= 1)
    accLoss += __shfl_xor(accLoss, off, 32);
  if (lane == 0)
    atomicAdd(out, accLoss * invB);
}

extern "C" void kernel_launch(void* const* d_in, const int* in_sizes, int n_in,
                              void* d_out, int out_size, void* d_ws, size_t ws_size,
                              hipStream_t stream) {
  (void)n_in; (void)out_size; (void)d_ws; (void)ws_size;
  const float* x1  = (const float*)d_in[0];
  const int*   tgt = (const int*)d_in[1];
  float*       out = (float*)d_out;

  const int nRows  = in_sizes[1];               // B = 4194304
  const int nPairs = (nRows + 31) / 32;         // 32-row units
  const float invB = 1.0f / (float)nRows;

  rbf_zero_out<<<1, 64, 0, stream>>>(out);

  const int threads = 256;                      // 8 wave32 per block
  int blocks = 2048;                            // 16384 waves, grid-stride over pairs
  if (blocks > nPairs) blocks = nPairs > 0 ? nPairs : 1;
  rbf_loss_wmma<<<blocks, threads, 0, stream>>>(x1, tgt, out, nRows, nPairs, invB);
}